// MT2VEncoderFusion_90469191123498
// MI455X (gfx1250) — compile-verified
//
#include <hip/hip_runtime.h>
#include <hip/hip_bf16.h>

// Problem constants from the reference.
#define BB 32
#define LL 7
#define MM 6
#define CC 3
#define HH 128
#define WW 128
#define TOPK 3
#define HW (HH * WW)          // 16384
#define NBL (BB * LL)         // 224 (b,l) pairs
#define NROWS (NBL * TOPK)    // 672 output rows, = 42 * 16 exactly
#define ROWTILES (NROWS / 16) // 42

typedef __attribute__((ext_vector_type(2))) float v2f;
typedef __attribute__((ext_vector_type(8))) float v8f;

// ---------------------------------------------------------------------------
// Kernel 1: per (b,l) top-3 selection + straight-through scale + source offset.
// Descriptor per output row r = bl*3 + k : { src byte offset of the selected
// channel-0 image, scale = (1 - w) + w }.
// jax.lax.top_k tie-break: equal values -> lower index first ("> " keeps the
// earlier max).
// ---------------------------------------------------------------------------
__global__ void mt2v_topk_desc_kernel(const float* __restrict__ wts,
                                      uint2* __restrict__ desc) {
    int bl = blockIdx.x * blockDim.x + threadIdx.x;
    if (bl >= NBL) return;

    float v[MM], vv[MM];
#pragma unroll
    for (int m = 0; m < MM; ++m) {
        v[m]  = wts[bl * MM + m];
        vv[m] = v[m];
    }
#pragma unroll
    for (int k = 0; k < TOPK; ++k) {
        int best = 0;
        float bv = vv[0];
#pragma unroll
        for (int m = 1; m < MM; ++m) {
            if (vv[m] > bv) { bv = vv[m]; best = m; }
        }
        float w0    = v[best];
        float scale = (1.0f - w0) + w0;          // straight-through scale
        // imgs[b,l,best,0,:,:] start, in bytes (fits in 32 bits: < 265 MB)
        unsigned off = ((unsigned)(bl * MM + best) * (unsigned)CC) *
                       (unsigned)HW * 4u;
        uint2 d;
        d.x = off;
        d.y = __float_as_uint(scale);
        desc[bl * TOPK + k] = d;
        vv[best] = -__builtin_inff();
    }
}

// ---------------------------------------------------------------------------
// Kernel 2: gather-scale via chained V_WMMA_F32_16X16X4_F32.
// D(16x16) = A(16x16 diagonal of scales, chained as 4 K=4 steps) x B(16
// gathered source rows x 16 columns).
//
// f32 operand layouts (ISA 7.12.2), reg r / half h (h = lane>>4):
//   A 16x4 : M = lane&15, K = r + 2h
//   B 4x16 : N = lane&15, K = r + 2h   (row-striped symmetric to A)
//   C/D    : N = lane&15, M = reg + 8h
// ---------------------------------------------------------------------------
__global__ void mt2v_gather_wmma_kernel(const float* __restrict__ imgs,
                                        const uint2* __restrict__ desc,
                                        float* __restrict__ out) {
    __shared__ uint2 sdesc[16];
    const int tid     = threadIdx.x;
    const int rowtile = blockIdx.y;

    if (tid < 16) sdesc[tid] = desc[rowtile * 16 + tid];
    __syncthreads();

    const int lane = tid & 31;
    const int wave = tid >> 5;
    const int n    = lane & 15;  // N column / A row within tile
    const int h    = lane >> 4;  // half-wave

    // Diagonal-A components and gathered row base offsets, per K-step j.
    const float s_m = __uint_as_float(sdesc[n].y);  // scale for row m = n
    float    ax[4], ay[4];
    unsigned off0[4], off1[4];
#pragma unroll
    for (int j = 0; j < 4; ++j) {
        const int r0 = 4 * j + 2 * h;               // K of reg0 at this step
        ax[j]   = (n == r0)     ? s_m : 0.0f;       // A[n][r0]
        ay[j]   = (n == r0 + 1) ? s_m : 0.0f;       // A[n][r0+1]
        off0[j] = sdesc[r0].x;
        off1[j] = sdesc[r0 + 1].x;
    }

    const char* __restrict__ base = (const char*)imgs;
    float* __restrict__ outp = out + (size_t)rowtile * 16 * HW;

#pragma unroll
    for (int t = 0; t < 4; ++t) {
        const int tile = blockIdx.x * 32 + wave * 4 + t;  // 1024 col tiles
        const int col  = tile * 16 + n;
        const unsigned coff = (unsigned)col << 2;

        // Gather B: 2 source-row elements per K-step.
        float bx[4], by[4];
#pragma unroll
        for (int j = 0; j < 4; ++j) {
            bx[j] = *(const float*)(base + off0[j] + coff);
            by[j] = *(const float*)(base + off1[j] + coff);
        }

        v8f c = {};
#pragma unroll
        for (int j = 0; j < 4; ++j) {
            v2f a, b;
            a.x = ax[j]; a.y = ay[j];
            b.x = bx[j]; b.y = by[j];
            c = __builtin_amdgcn_wmma_f32_16x16x4_f32(
                    /*neg_a=*/false, a, /*neg_b=*/false, b,
                    /*c_mod=*/(short)0, c,
                    /*reuse_a=*/false, /*reuse_b=*/false);
        }

        // Scatter D: VGPR v holds rows M = v (lanes 0-15) and v+8 (16-31).
#pragma unroll
        for (int v = 0; v < 8; ++v) {
            const int row = v + 8 * h;
            outp[(size_t)row * HW + col] = c[v];
        }
    }
}

extern "C" void kernel_launch(void* const* d_in, const int* in_sizes, int n_in,
                              void* d_out, int out_size, void* d_ws, size_t ws_size,
                              hipStream_t stream) {
    const float* imgs = (const float*)d_in[0];
    const float* wts  = (const float*)d_in[1];
    float* out        = (float*)d_out;
    uint2* desc       = (uint2*)d_ws;  // 672 * 8 = 5376 bytes of scratch

    (void)in_sizes; (void)n_in; (void)out_size; (void)ws_size;

    mt2v_topk_desc_kernel<<<dim3(1), dim3(256), 0, stream>>>(wts, desc);

    // 1024 column tiles of 16; block = 8 waves x 4 tiles -> grid.x = 32.
    // grid.y = 42 row tiles of 16 output rows.
    mt2v_gather_wmma_kernel<<<dim3(32, ROWTILES), dim3(256), 0, stream>>>(
        imgs, desc, out);
}